// GraphConvolution_1297080124150
// MI455X (gfx1250) — compile-verified
//
#include <hip/hip_runtime.h>
#include <hip/hip_bf16.h>

#define IN_DIM 256
#define OUT_DIM 64
#define A_PAD 260  // 256 + 4 floats pad -> conflict-free ds_load_b64 A fragments

typedef float v2f __attribute__((ext_vector_type(2)));
typedef float v4f __attribute__((ext_vector_type(4)));
typedef float v8f __attribute__((ext_vector_type(8)));

// ---------------------------------------------------------------------------
// Zero the accumulator (segment_sum target) -- re-run every call since the
// harness poisons d_out.
// ---------------------------------------------------------------------------
__global__ void gcn_zero_kernel(float* __restrict__ out, int n) {
    int i = blockIdx.x * blockDim.x + threadIdx.x;
    if (i < n) out[i] = 0.0f;
}

// ---------------------------------------------------------------------------
// xw = x @ W using V_WMMA_F32_16X16X4_F32 (exact fp32).
// Block = 128 threads = 4 waves. Block owns a 16-row tile of x (staged in
// LDS with padded rows -> conflict-free ds_load_b64 A fragments). Wave w
// owns output columns [16w, 16w+16). K loop: 64 steps of K=4.
// ---------------------------------------------------------------------------
__global__ __launch_bounds__(128) void gcn_gemm_kernel(
        const float* __restrict__ x, const float* __restrict__ w,
        float* __restrict__ xw, int n_nodes) {
    __shared__ float sA[16 * A_PAD];

    const int tid  = threadIdx.x;
    const int lane = tid & 31;
    const int wav  = tid >> 5;          // 0..3 -> n-tile
    const int row0 = blockIdx.x * 16;
    const bool full_tile = (row0 + 16 <= n_nodes);

    // Cooperative, coalesced load of 16x256 fp32 x-tile into LDS.
    if (full_tile) {
        for (int i = tid; i < 16 * (IN_DIM / 4); i += 128) {
            int r  = i >> 6;            // / (IN_DIM/4)
            int c4 = i & 63;
            *(v4f*)(&sA[r * A_PAD + c4 * 4]) =
                *(const v4f*)(x + (size_t)(row0 + r) * IN_DIM + c4 * 4);
        }
    } else {
        for (int i = tid; i < 16 * (IN_DIM / 4); i += 128) {
            int r  = i >> 6;
            int c4 = i & 63;
            v4f v = {};
            if (row0 + r < n_nodes)
                v = *(const v4f*)(x + (size_t)(row0 + r) * IN_DIM + c4 * 4);
            *(v4f*)(&sA[r * A_PAD + c4 * 4]) = v;
        }
    }
    __syncthreads();

    const int hlf = lane >> 4;          // 0: K pair {k0,k0+1}, 1: {k0+2,k0+3}
    const int l   = lane & 15;          // M (A) / N (B,C,D) index
    const int n0  = wav * 16;

    v8f acc = {};

    #pragma unroll 4
    for (int k0 = 0; k0 < IN_DIM; k0 += 4) {
        const int ka = k0 + hlf * 2;
        // A fragment (16x4 f32): lane<16 -> A[l][k0..k0+1], lane>=16 -> A[l][k0+2..k0+3]
        v2f a = *(const v2f*)(&sA[l * A_PAD + ka]);
        // B fragment (4x16 f32): rows striped across lane halves, mirroring A.
        v2f b;
        b.x = w[ka * OUT_DIM + n0 + l];
        b.y = w[(ka + 1) * OUT_DIM + n0 + l];
        acc = __builtin_amdgcn_wmma_f32_16x16x4_f32(
            /*neg_a=*/false, a, /*neg_b=*/false, b,
            /*c_mod=*/(short)0, acc, /*reuse_a=*/false, /*reuse_b=*/false);
    }

    // D layout: VGPR r -> M = r + 8*hlf, N = n0 + l.
    // Uniform fast path: one base pointer + 8 immediate-offset stores
    // (two 64B segments per store, compiler can clause them).
    float* dst = xw + (size_t)(row0 + hlf * 8) * OUT_DIM + n0 + l;
    if (full_tile) {
        #pragma unroll
        for (int r = 0; r < 8; ++r)
            dst[r * OUT_DIM] = acc[r];
    } else {
        #pragma unroll
        for (int r = 0; r < 8; ++r)
            if (row0 + hlf * 8 + r < n_nodes)
                dst[r * OUT_DIM] = acc[r];
    }
}

// ---------------------------------------------------------------------------
// Edge scatter: out[dst] += w * xw[src]. 16 lanes per edge, float4 per lane
// (coalesced 256B gather, mostly L2-resident), 4 f32 atomics per lane
// (no-return -> STOREcnt, resolved in L2 atomic units).
// ---------------------------------------------------------------------------
__global__ void gcn_scatter_kernel(
        const float* __restrict__ xw, const int* __restrict__ esrc,
        const int* __restrict__ edst, const float* __restrict__ ew,
        float* out, int n_edges) {
    long long gid = (long long)blockIdx.x * blockDim.x + threadIdx.x;
    int e = (int)(gid >> 4);
    int q = (int)(gid & 15);
    if (e >= n_edges) return;

    int   src = esrc[e];
    int   dst = edst[e];
    float wgt = ew[e];

    v4f v = *(const v4f*)(xw + (size_t)src * OUT_DIM + q * 4);
    float* o = out + (size_t)dst * OUT_DIM + q * 4;
    atomicAdd(o + 0, wgt * v.x);
    atomicAdd(o + 1, wgt * v.y);
    atomicAdd(o + 2, wgt * v.z);
    atomicAdd(o + 3, wgt * v.w);
}

// ---------------------------------------------------------------------------
// Final activation.
// ---------------------------------------------------------------------------
__global__ void gcn_relu_kernel(float* __restrict__ out, int n) {
    int i = blockIdx.x * blockDim.x + threadIdx.x;
    if (i < n) out[i] = fmaxf(out[i], 0.0f);
}

extern "C" void kernel_launch(void* const* d_in, const int* in_sizes, int n_in,
                              void* d_out, int out_size, void* d_ws, size_t ws_size,
                              hipStream_t stream) {
    const float* x    = (const float*)d_in[0];
    const float* w    = (const float*)d_in[1];
    const int*   esrc = (const int*)d_in[2];
    const int*   edst = (const int*)d_in[3];
    const float* ew   = (const float*)d_in[4];
    float* out = (float*)d_out;
    float* xw  = (float*)d_ws;            // n_nodes * 64 fp32 = 25.6 MB

    const int n_nodes = in_sizes[0] / IN_DIM;
    const int n_edges = in_sizes[2];
    const int n_out   = n_nodes * OUT_DIM;

    gcn_zero_kernel<<<(n_out + 255) / 256, 256, 0, stream>>>(out, n_out);

    gcn_gemm_kernel<<<(n_nodes + 15) / 16, 128, 0, stream>>>(x, w, xw, n_nodes);

    long long sthreads = (long long)n_edges * 16;
    gcn_scatter_kernel<<<(unsigned)((sthreads + 255) / 256), 256, 0, stream>>>(
        xw, esrc, edst, ew, out, n_edges);

    gcn_relu_kernel<<<(n_out + 255) / 256, 256, 0, stream>>>(out, n_out);
}